// GCCLoss_79955111182640
// MI455X (gfx1250) — compile-verified
//
#include <hip/hip_runtime.h>

// ---------------------------------------------------------------------------
// Chamfer distance on MI455X (gfx1250), wave32.
//   d2[b][i][j] = max(0, ||gt_i||^2 + ||pred_j||^2 - 2 gt_i . pred_j)
//   loss = mean_i min_j d2 + mean_j min_i d2
//
// V_WMMA_F32_16X16X4_F32 computes the ENTIRE squared distance:
//   A row i = [-2x0, -2x1, -2x2, ||x||^2]   (K padded 3 -> 4)
//   B col j = [y0, y1, y2, 1]
//   C       = broadcast(||y_j||^2)   (D layout: every VGPR of lane L is col L%16)
//   => D[i][j] = ||x_i||^2 - 2 x_i.y_j + ||y_j||^2   straight out of the MMA.
// Clamp max(0,.) is monotone, so it is deferred past the min-reductions.
// dist2 combined across workgroups with global_atomic_min_u32 (f32 bit order
// == u32 order for non-negative values) -- order-independent => deterministic.
// ---------------------------------------------------------------------------

typedef __attribute__((ext_vector_type(2))) float v2f;
typedef __attribute__((ext_vector_type(8))) float v8f;

#define BATCH          16
#define NPTS           4096          // N == M
#define CHUNKS         8             // i-chunks per batch -> 16*8 = 128 WGs
#define WAVES_PER_WG   8             // 256 threads
#define TILES_PER_WAVE 4             // 4 x 16 rows = 64 rows per wave
#define ROWS_PER_WG    (NPTS / CHUNKS)                // 512
#define ROWS_PER_WAVE  (ROWS_PER_WG / WAVES_PER_WG)   // 64
#define JTILE          16
#define NPARTIALS      (BATCH * CHUNKS * WAVES_PER_WG) // 1024

// ---------------------------------------------------------------- init ------
__global__ __launch_bounds__(256) void chamfer_init(unsigned* __restrict__ dist2u) {
    int i = blockIdx.x * 256 + threadIdx.x;
    if (i < BATCH * NPTS) dist2u[i] = 0x7F800000u;   // +inf bit pattern
}

// ---------------------------------------------------------------- main ------
__global__ __launch_bounds__(256) void chamfer_main(const float* __restrict__ gt,
                                                    const float* __restrict__ pred,
                                                    unsigned* __restrict__ dist2u,
                                                    float* __restrict__ partials)
{
    const int lane  = threadIdx.x & 31;
    const int wave  = threadIdx.x >> 5;
    const int half  = lane >> 4;      // 0: K=0,1 ; 1: K=2,3 (A/B frag halves)
    const int m     = lane & 15;      // row (A) / col (B/C/D) index within tile
    const int chunk = blockIdx.x;     // i-chunk within batch
    const int b     = blockIdx.y;     // batch

    const float* __restrict__ gtb = gt   + (size_t)b * NPTS * 3;
    const float* __restrict__ prb = pred + (size_t)b * NPTS * 3;
    unsigned* __restrict__ d2row  = dist2u + (size_t)b * NPTS;

    // --- A fragments for this wave's 4 row-tiles (loaded once) -------------
    // A-matrix 16x4 f32 layout: lanes 0-15 hold {K0,K1}, lanes 16-31 {K2,K3}.
    v2f afrag[TILES_PER_WAVE];
#pragma unroll
    for (int t = 0; t < TILES_PER_WAVE; ++t) {
        int row = chunk * ROWS_PER_WG + wave * ROWS_PER_WAVE + t * 16 + m;
        const float* p = gtb + (size_t)row * 3;
        float x0 = p[0], x1 = p[1], x2 = p[2];
        float xx = x0 * x0 + x1 * x1 + x2 * x2;
        v2f a;
        a.x = half ? (-2.0f * x2) : (-2.0f * x0);
        a.y = half ? xx           : (-2.0f * x1);
        afrag[t] = a;
    }

    // Running per-lane mins for dist1 (rows), UNclamped; clamp deferred.
    float d1acc[TILES_PER_WAVE][8];
#pragma unroll
    for (int t = 0; t < TILES_PER_WAVE; ++t)
#pragma unroll
        for (int r = 0; r < 8; ++r) d1acc[t][r] = __builtin_inff();

    // --- Sweep all pred tiles ---------------------------------------------
#pragma unroll 2
    for (int j0 = 0; j0 < NPTS; j0 += JTILE) {
        const float* q = prb + (size_t)(j0 + m) * 3;
        float y0 = q[0], y1 = q[1], y2 = q[2];
        float yy = y0 * y0 + y1 * y1 + y2 * y2;   // ||y_{j0+m}||^2

        // Speculative prefetch of the next tile (dropped if OOB on last iter).
        __builtin_prefetch(prb + (size_t)(j0 + JTILE + m) * 3, 0, 1);

        // B-matrix 4x16 f32: lanes 0-15 hold {K0,K1}, lanes 16-31 {K2,K3}.
        v2f bfrag;
        bfrag.x = half ? y2   : y0;
        bfrag.y = half ? 1.0f : y1;

        // C = per-lane broadcast of yy -> WMMA emits the full squared dist.
        v8f cinit;
#pragma unroll
        for (int r = 0; r < 8; ++r) cinit[r] = yy;

        float d2min = __builtin_inff();
#pragma unroll
        for (int t = 0; t < TILES_PER_WAVE; ++t) {
            // (neg_a, A, neg_b, B, c_mod, C, reuse_a, reuse_b)
            v8f d = __builtin_amdgcn_wmma_f32_16x16x4_f32(
                        false, afrag[t], false, bfrag, (short)0, cinit,
                        false, false);
#pragma unroll
            for (int r = 0; r < 8; ++r) {
                d1acc[t][r] = fminf(d1acc[t][r], d[r]);  // min over j (running)
                d2min       = fminf(d2min, d[r]);        // min over tile rows
            }
        }
        // Lane L holds the min over rows {r} (half 0) / {r+8} (half 1) of all
        // 4 tiles for column n = L&15. Both halves hit the same dist2 slot;
        // the hardware atomic-min merges them (no shuffle, no EXEC juggling).
        d2min = fmaxf(d2min, 0.0f);                      // clamp before u32 trick
        atomicMin(&d2row[j0 + m], __float_as_uint(d2min));
    }

    // --- dist1: clamp, cross-lane min within each 16-lane half, wave sum ---
    float laneSum = 0.0f;
#pragma unroll
    for (int t = 0; t < TILES_PER_WAVE; ++t)
#pragma unroll
        for (int r = 0; r < 8; ++r) {
            float v = fmaxf(d1acc[t][r], 0.0f);          // deferred clamp
            v = fminf(v, __shfl_xor(v, 1, 32));
            v = fminf(v, __shfl_xor(v, 2, 32));
            v = fminf(v, __shfl_xor(v, 4, 32));
            v = fminf(v, __shfl_xor(v, 8, 32));  // all lanes of half: row min
            laneSum += v;
        }
    // Each distinct row-min is replicated across 16 lanes; full-wave add then
    // divide by 16 gives the sum over this wave's 64 rows.
    laneSum += __shfl_xor(laneSum, 1, 32);
    laneSum += __shfl_xor(laneSum, 2, 32);
    laneSum += __shfl_xor(laneSum, 4, 32);
    laneSum += __shfl_xor(laneSum, 8, 32);
    laneSum += __shfl_xor(laneSum, 16, 32);
    if (lane == 0) {
        int wg = (b * CHUNKS + chunk) * WAVES_PER_WG + wave;
        partials[wg] = laneSum * (1.0f / 16.0f);
    }
}

// ---------------------------------------------------------------- final -----
__global__ __launch_bounds__(1024) void chamfer_final(const unsigned* __restrict__ dist2u,
                                                      const float* __restrict__ partials,
                                                      float* __restrict__ out)
{
    __shared__ float red[1024];
    int tid = threadIdx.x;

    float s2 = 0.0f;
    for (int i = tid; i < BATCH * NPTS; i += 1024)
        s2 += __uint_as_float(dist2u[i]);
    float s1 = partials[tid];                       // exactly 1024 partials

    red[tid] = (s1 + s2) * (1.0f / (float)(BATCH * NPTS));
    __syncthreads();
    for (int off = 512; off > 0; off >>= 1) {
        if (tid < off) red[tid] += red[tid + off];
        __syncthreads();
    }
    if (tid == 0) out[0] = red[0];
}

// ---------------------------------------------------------------- launch ----
extern "C" void kernel_launch(void* const* d_in, const int* in_sizes, int n_in,
                              void* d_out, int out_size, void* d_ws, size_t ws_size,
                              hipStream_t stream)
{
    const float* gt   = (const float*)d_in[0];   // (16,4096,3) f32
    const float* pred = (const float*)d_in[1];   // (16,4096,3) f32
    float* out        = (float*)d_out;           // scalar f32

    unsigned* dist2u = (unsigned*)d_ws;                                   // 256 KB
    float* partials  = (float*)((char*)d_ws + (size_t)BATCH * NPTS * 4);  // 4 KB

    chamfer_init<<<(BATCH * NPTS + 255) / 256, 256, 0, stream>>>(dist2u);

    dim3 grid(CHUNKS, BATCH);
    chamfer_main<<<grid, 256, 0, stream>>>(gt, pred, dist2u, partials);

    chamfer_final<<<1, 1024, 0, stream>>>(dist2u, partials, out);
}